// DiceLoss_61589831025000
// MI455X (gfx1250) — compile-verified
//
#include <hip/hip_runtime.h>
#include <stdint.h>
#include <stddef.h>

// DiceLoss: N=16, C=8, H=W=512.
// Memory-bound: 151 MB / 23.3 TB/s ~= 6.5us floor. Strategy: stream y_pred once
// via CDNA5 async global->LDS B128 copies, double-buffered (ping-pong LDS tiles,
// s_wait_asynccnt 8 overlap). Packed 8x8-bit per-lane count accumulators, wave
// shfl reduce, hierarchical atomics into 24 u32 counters in d_ws, then a
// 1-block finalize kernel computes the scalar dice mean.

#define EPS_F 1e-5f
#define HW_   (512u * 512u)          // pixels per plane (2^18)
#define NPIX_ (16u * HW_)            // total pixels
#define NGRP_ (NPIX_ / 4u)           // float4 groups
#define NBLK  512
#define NTPB  256
#define NITER 8                      // NGRP_ / (NBLK*NTPB), exact

struct alignas(16) F4 { float v[4]; };
struct alignas(16) I4 { int   v[4]; };

#if defined(__AMDGCN__) && defined(__gfx1250__) && \
    __has_builtin(__builtin_amdgcn_global_load_async_to_lds_b128)
#define USE_ASYNC 1
#else
#define USE_ASYNC 0
#endif

#if USE_ASYNC
typedef int v4i __attribute__((vector_size(16)));
typedef __attribute__((address_space(1))) v4i* gptr_t;   // global (64-bit)
typedef __attribute__((address_space(3))) v4i* lptr_t;   // LDS (32-bit)

__device__ __forceinline__ void async_ld16(const float* g, const void* l) {
  // GLOBAL_LOAD_ASYNC_TO_LDS_B128: 16B memory -> LDS per lane, tracked by
  // ASYNCcnt. Generic LDS pointer low 32 bits == LDS byte address (ISA
  // aperture rule), so reinterpret via integers instead of addrspacecast.
  __builtin_amdgcn_global_load_async_to_lds_b128(
      (gptr_t)(uintptr_t)g,
      (lptr_t)(uint32_t)(uintptr_t)l,
      0 /*imm offset*/, 0 /*imm cpol*/);
}

template <int N>
__device__ __forceinline__ void wait_async() {
#if __has_builtin(__builtin_amdgcn_s_wait_asynccnt)
  __builtin_amdgcn_s_wait_asynccnt(N);
#else
  asm volatile("s_wait_asynccnt %0" ::"i"(N) : "memory");
#endif
}
#endif

__global__ __launch_bounds__(NTPB)
void dice_count(const int* __restrict__ yt, const float* __restrict__ yp,
                uint32_t* __restrict__ cnt) {
#if USE_ASYNC
  __shared__ F4 tile[2][8][NTPB];    // 2 x 32 KB ping-pong async landing pads
#endif
  __shared__ uint32_t bsum[24];
  if (threadIdx.x < 24u) bsum[threadIdx.x] = 0u;
  __syncthreads();

  const uint32_t tid    = blockIdx.x * NTPB + threadIdx.x;
  const uint32_t stride = (uint32_t)NBLK * NTPB;

  // Packed per-lane counters: byte c of pa/ga/ta = count for class c.
  // NITER iters * 4 pixels => max 32 per byte, no overflow.
  uint64_t pa = 0, ga = 0, ta = 0;

#if USE_ASYNC
  // Issue the 8 class-plane copies for group-index g into buffer buf.
  auto issue = [&](int buf, uint32_t g) {
    uint32_t p  = g << 2;
    uint32_t n  = p >> 18;            // image index (HW = 2^18)
    uint32_t hw = p & (HW_ - 1u);
    const float* base = yp + (size_t)n * (size_t)(8u * HW_) + hw;
#pragma unroll
    for (int c = 0; c < 8; ++c)
      async_ld16(base + (size_t)c * HW_, &tile[buf][c][threadIdx.x]);
  };

  issue(0, tid);                      // prologue
  for (int it = 0; it < NITER; ++it) {
    uint32_t p = (tid + (uint32_t)it * stride) << 2;
    if (it + 1 < NITER) {
      issue((it + 1) & 1, tid + (uint32_t)(it + 1) * stride);
      // Async loads complete in order: <=8 outstanding means the current
      // tile's 8 copies have landed while the next 8 stay in flight.
      wait_async<8>();
    } else {
      wait_async<0>();
    }

    I4 gt4 = *(const I4*)(yt + p);    // global_load_b128
    F4 r[8];
#pragma unroll
    for (int c = 0; c < 8; ++c) r[c] = tile[it & 1][c][threadIdx.x]; // ds_load_b128

#pragma unroll
    for (int j = 0; j < 4; ++j) {
      int best = 0; float bv = r[0].v[j];
#pragma unroll
      for (int c = 1; c < 8; ++c) {   // strict '>' keeps lowest index on ties
        if (r[c].v[j] > bv) { bv = r[c].v[j]; best = c; }
      }
      int gt = gt4.v[j] & 7;
      pa += 1ull << (best << 3);
      ga += 1ull << (gt << 3);
      if (best == gt) ta += 1ull << (gt << 3);
    }
  }
#else
  for (int it = 0; it < NITER; ++it) {
    uint32_t g  = tid + (uint32_t)it * stride;
    uint32_t p  = g << 2;
    uint32_t n  = p >> 18;
    uint32_t hw = p & (HW_ - 1u);
    const float* base = yp + (size_t)n * (size_t)(8u * HW_) + hw;
    I4 gt4 = *(const I4*)(yt + p);
    F4 r[8];
#pragma unroll
    for (int c = 0; c < 8; ++c) r[c] = *(const F4*)(base + (size_t)c * HW_);
#pragma unroll
    for (int j = 0; j < 4; ++j) {
      int best = 0; float bv = r[0].v[j];
#pragma unroll
      for (int c = 1; c < 8; ++c) {
        if (r[c].v[j] > bv) { bv = r[c].v[j]; best = c; }
      }
      int gt = gt4.v[j] & 7;
      pa += 1ull << (best << 3);
      ga += 1ull << (gt << 3);
      if (best == gt) ta += 1ull << (gt << 3);
    }
  }
#endif

  // Unpack 3x8 byte counters -> 24 u32, wave32 shfl reduction, then LDS+global.
  uint32_t c24[24];
#pragma unroll
  for (int c = 0; c < 8; ++c) {
    c24[c]      = (uint32_t)(pa >> (c << 3)) & 0xFFu;
    c24[8 + c]  = (uint32_t)(ga >> (c << 3)) & 0xFFu;
    c24[16 + c] = (uint32_t)(ta >> (c << 3)) & 0xFFu;
  }
#pragma unroll
  for (int k = 0; k < 24; ++k) {
    uint32_t x = c24[k];
#pragma unroll
    for (int off = 16; off > 0; off >>= 1)
      x += (uint32_t)__shfl_down((int)x, (unsigned)off, 32);
    c24[k] = x;
  }
  if ((threadIdx.x & 31u) == 0u) {
#pragma unroll
    for (int k = 0; k < 24; ++k) atomicAdd(&bsum[k], c24[k]);
  }
  __syncthreads();
  if (threadIdx.x < 24u) atomicAdd(&cnt[threadIdx.x], bsum[threadIdx.x]);
}

__global__ void dice_zero24(uint32_t* __restrict__ cnt) {
  if (threadIdx.x < 24u) cnt[threadIdx.x] = 0u;
}

__global__ void dice_finalize(const uint32_t* __restrict__ cnt,
                              float* __restrict__ out) {
  if (threadIdx.x == 0u && blockIdx.x == 0u) {
    float s = 0.0f;
#pragma unroll
    for (int c = 1; c < 8; ++c) {                 // class 0 dropped
      float pc = (float)cnt[c];
      float gc = (float)cnt[8 + c];
      float tp = (float)cnt[16 + c];
      // 2tp + fp + fn = 2tp + (pred-tp) + (gt-tp) = pred + gt
      s += (2.0f * tp + EPS_F) / (pc + gc + EPS_F);
    }
    out[0] = s * (1.0f / 7.0f);
  }
}

extern "C" void kernel_launch(void* const* d_in, const int* in_sizes, int n_in,
                              void* d_out, int out_size, void* d_ws, size_t ws_size,
                              hipStream_t stream) {
  (void)in_sizes; (void)n_in; (void)out_size; (void)ws_size;
  const int*   yt  = (const int*)d_in[0];    // y_true: (16,512,512) int32
  const float* yp  = (const float*)d_in[1];  // y_pred: (16,8,512,512) float32
  uint32_t*    cnt = (uint32_t*)d_ws;        // 24 u32 counters
  float*       out = (float*)d_out;

  hipLaunchKernelGGL(dice_zero24,   dim3(1),    dim3(32),   0, stream, cnt);
  hipLaunchKernelGGL(dice_count,    dim3(NBLK), dim3(NTPB), 0, stream, yt, yp, cnt);
  hipLaunchKernelGGL(dice_finalize, dim3(1),    dim3(32),   0, stream, cnt, out);
}